// SSMStateCompressor_68032281969132
// MI455X (gfx1250) — compile-verified
//
#include <hip/hip_runtime.h>
#include <math.h>

typedef __attribute__((ext_vector_type(2))) float v2f;
typedef __attribute__((ext_vector_type(8))) float v8f;

#define FLAT 32768          // 2048 * 16
#define NTHREADS 256
#define PADDED 34816        // padi(32767)+1 = 34812, rounded up
// CLIP = 4/sqrt(32768); DELTA = 2*CLIP/15
#define F_CLIP       0.02209708691207961f
#define F_DELTA      0.002946278254943948f
#define F_INV_DELTA  339.4112549695428f
#define F_INV_SQRT_D 0.005524271728019903f

// Padded LDS layout: 4 pad dwords per 64. Multiple-of-4 pad keeps every
// 4-dword-aligned group contiguous and 16B-aligned (b64/b128 legal), while
// stride-16 lane patterns land on banks 16*(n&3) + 4*(n>>2) + c -> 16 distinct
// banks, and stride-1 patterns stay conflict-free.
__device__ __forceinline__ int padi(int i) { return i + ((i >> 6) << 2); }

// H16 Sylvester (bit-parity) entry: (-1)^popc(m&k)
__device__ __forceinline__ float hadH16(int m, int k) {
    return (__popc(m & k) & 1) ? -1.0f : 1.0f;
}
// blockdiag(H8,H8) = I2 (x) H8
__device__ __forceinline__ float hadBlk8(int m, int k) {
    if ((m ^ k) & 8) return 0.0f;
    return hadH16(m & 7, k & 7);
}

// Logical element index for tile t, transform-axis `row`, free-axis `col`.
// Stage 0: rows = bits 3:0   Stage 1: rows = bits 7:4
// Stage 2: rows = bits 11:8  Stage 3: rows = (bit11, bits14:12)
template <int S>
__device__ __forceinline__ int stage_addr(int t, int row, int col) {
    if (S == 0) return t * 256 + col * 16 + row;
    if (S == 1) return t * 256 + row * 16 + col;
    if (S == 2) return (t >> 4) * 4096 + row * 256 + col * 16 + (t & 15);
    /* S == 3 */
    return (row & 7) * 4096 + (row >> 3) * 2048 + t * 16 + col;
}

// One FWHT stage over the row held in (padded) LDS, in place.
// Each wave owns 16 disjoint 16x16 tiles: D = A(16x16) * X(16x16) via
// 4 chained V_WMMA_F32_16X16X4_F32. A is prebuilt in VGPRs with any scalar
// factors (1/scale, 1/sqrt(d)) folded in. QUANT folds 4-bit quant+dequant
// into the writeback.
template <int S, bool QUANT>
__device__ __forceinline__ void stage(float* lds, int wave, int lane, const v2f* A) {
    const bool hi = lane >= 16;
    const int n = lane & 15;               // B/D column
    const int kbase = hi ? 2 : 0;          // K-slot assignment per f32 WMMA layout

#pragma unroll 4
    for (int ti = 0; ti < 16; ++ti) {
        const int t = wave * 16 + ti;
        v8f acc = {};
#pragma unroll
        for (int kb = 0; kb < 4; ++kb) {
            const int k0 = kb * 4 + kbase;
            v2f B;
            if (S == 0) {
                // rows are memory-contiguous: one 8B-aligned vector load
                B = *(const v2f*)(lds + padi(stage_addr<S>(t, k0, n)));
            } else {
                B[0] = lds[padi(stage_addr<S>(t, k0, n))];
                B[1] = lds[padi(stage_addr<S>(t, k0 + 1, n))];
            }
            acc = __builtin_amdgcn_wmma_f32_16x16x4_f32(
                false, A[kb], false, B, (short)0, acc, false, false);
        }
        if (QUANT) {
#pragma unroll
            for (int g = 0; g < 8; ++g) {
                float q = rintf((acc[g] + F_CLIP) * F_INV_DELTA); // round-half-even
                q = fminf(fmaxf(q, 0.0f), 15.0f);
                acc[g] = q * F_DELTA - F_CLIP;
            }
        }
        // D layout: VGPR g -> row M = g + (lane<16 ? 0 : 8), col N = lane&15
        if (S == 0) {
            const int m0 = hi ? 8 : 0;     // rows memory-contiguous: b128 stores
            float4 w0, w1;
            w0.x = acc[0]; w0.y = acc[1]; w0.z = acc[2]; w0.w = acc[3];
            w1.x = acc[4]; w1.y = acc[5]; w1.z = acc[6]; w1.w = acc[7];
            *(float4*)(lds + padi(stage_addr<0>(t, m0,     n))) = w0;
            *(float4*)(lds + padi(stage_addr<0>(t, m0 + 4, n))) = w1;
        } else {
#pragma unroll
            for (int g = 0; g < 8; ++g) {
                const int m = g + (hi ? 8 : 0);
                lds[padi(stage_addr<S>(t, m, n))] = acc[g];
            }
        }
    }
}

__global__ void __launch_bounds__(NTHREADS)
ssm_compress_roundtrip(const float* __restrict__ h,
                       const float* __restrict__ signs,
                       float* __restrict__ out) {
    extern __shared__ __align__(16) float lds[];   // PADDED floats + 256 reduction
    float* red = lds + PADDED;

    const int row  = blockIdx.x;
    const int tid  = threadIdx.x;
    const int lane = tid & 31;
    const int wave = tid >> 5;

    const float4* __restrict__ h4 = (const float4*)h + (size_t)row * (FLAT / 4);
    const float4* __restrict__ s4 = (const float4*)signs;
    float4* __restrict__ o4       = (float4*)out + (size_t)row * (FLAT / 4);

    // ---- Prebuild WMMA A operands once (16x4 f32 layout:
    //      lane<16 holds K={k0,k0+1}, lane>=16 holds K={k0+2,k0+3}) ----
    const int m_a = lane & 15;
    const int kbase = (lane >= 16) ? 2 : 0;
    v2f Ah[4], Ab[4];
#pragma unroll
    for (int kb = 0; kb < 4; ++kb) {
        const int k0 = kb * 4 + kbase;
        Ah[kb][0] = hadH16(m_a, k0);
        Ah[kb][1] = hadH16(m_a, k0 + 1);
        Ab[kb][0] = hadBlk8(m_a, k0);
        Ab[kb][1] = hadBlk8(m_a, k0 + 1);
    }

    // ---- Load row, apply signs, accumulate sum of squares (|x| == |x*signs|) ----
    float ss = 0.0f;
#pragma unroll 4
    for (int it = 0; it < FLAT / 4 / NTHREADS; ++it) {
        const int i4 = it * NTHREADS + tid;
        const int pi = i4 + ((i4 >> 4) << 2);       // padded float4 slot
        const float4 v = h4[i4];
        const float4 s = s4[i4];
        ss += v.x * v.x + v.y * v.y + v.z * v.z + v.w * v.w;
        float4 w;
        w.x = v.x * s.x; w.y = v.y * s.y; w.z = v.z * s.z; w.w = v.w * s.w;
        *(float4*)(lds + 4 * pi) = w;
    }

    // ---- Deterministic tree reduction for the norm ----
    red[tid] = ss;
    __syncthreads();
    for (int off = NTHREADS / 2; off > 0; off >>= 1) {
        if (tid < off) red[tid] += red[tid + off];
        __syncthreads();
    }
    const float scale = sqrtf(red[0]);
    const float inv   = 1.0f / fmaxf(scale, 1e-8f);

    // ---- Scaled A variants: 1/scale into pass-1 stage 0,
    //      orthonormal 1/sqrt(d) into pass-1 stage 3 (before quantize) ----
    v2f Ah_inv[4], Ab_q[4];
#pragma unroll
    for (int kb = 0; kb < 4; ++kb) {
        Ah_inv[kb][0] = Ah[kb][0] * inv;
        Ah_inv[kb][1] = Ah[kb][1] * inv;
        Ab_q[kb][0]   = Ab[kb][0] * F_INV_SQRT_D;
        Ab_q[kb][1]   = Ab[kb][1] * F_INV_SQRT_D;
    }

    // ---- FWHT #1 + quantize/dequantize ----
    stage<0, false>(lds, wave, lane, Ah_inv);
    __syncthreads();
    stage<1, false>(lds, wave, lane, Ah);
    __syncthreads();
    stage<2, false>(lds, wave, lane, Ah);
    __syncthreads();
    stage<3, true >(lds, wave, lane, Ab_q);
    __syncthreads();

    // ---- FWHT #2 (its 1/sqrt(d) and *scale*signs folded into the store pass) ----
    stage<0, false>(lds, wave, lane, Ah);
    __syncthreads();
    stage<1, false>(lds, wave, lane, Ah);
    __syncthreads();
    stage<2, false>(lds, wave, lane, Ah);
    __syncthreads();
    stage<3, false>(lds, wave, lane, Ab);
    __syncthreads();

    const float fin = scale * F_INV_SQRT_D;
#pragma unroll 4
    for (int it = 0; it < FLAT / 4 / NTHREADS; ++it) {
        const int i4 = it * NTHREADS + tid;
        const int pi = i4 + ((i4 >> 4) << 2);
        const float4 v = *(const float4*)(lds + 4 * pi);
        const float4 s = s4[i4];
        float4 w;
        w.x = v.x * s.x * fin; w.y = v.y * s.y * fin;
        w.z = v.z * s.z * fin; w.w = v.w * s.w * fin;
        o4[i4] = w;
    }
}

extern "C" void kernel_launch(void* const* d_in, const int* in_sizes, int n_in,
                              void* d_out, int out_size, void* d_ws, size_t ws_size,
                              hipStream_t stream) {
    const float* h     = (const float*)d_in[0];
    const float* signs = (const float*)d_in[1];
    float* out         = (float*)d_out;

    const int B = in_sizes[0] / FLAT;                          // 1024 rows
    const size_t shmem = (PADDED + NTHREADS) * sizeof(float);  // 140288 bytes

    ssm_compress_roundtrip<<<dim3(B), dim3(NTHREADS), shmem, stream>>>(h, signs, out);
}